// LinkMessagePassing_86397562127195
// MI455X (gfx1250) — compile-verified
//
#include <hip/hip_runtime.h>

// LinkMessagePassing: out[dst[e]] += x[src[e]] for 1.6M edges, D=32 f32 features.
// edge_neighbors is (2, E) int64, row 0 = destination (segment id), row 1 = source.
//
// Roofline (MI455X): x (12.8 MB) and out (12.8 MB) are L2-resident (192 MB L2).
// Unique HBM traffic ~= int64 indices (25.6 MB) + first touch (~26 MB) -> ~2.2 us
// floor at 23.3 TB/s. Practical limiter: E*D = 51.2M f32 atomic-adds at L2.
// Design: 8 lanes/edge, one global_load_b128 per lane (contiguous 128 B per edge),
// native global_atomic_add_f32, and CDNA5 async global->LDS staging of the int64
// edge indices (ASYNCcnt / s_wait_asynccnt path).

#define D_FEAT 32
#define EDGES_PER_BLOCK 32
#define BLOCK_THREADS 256   // 8 lanes per edge * 32 edges

__global__ __launch_bounds__(BLOCK_THREADS) void scatter_add_kernel(
    const float* __restrict__ x,
    const long long* __restrict__ e_dst,   // edge_neighbors[0]
    const long long* __restrict__ e_src,   // edge_neighbors[1]
    float* __restrict__ out,
    int nEdges) {
    __shared__ long long s_dst[EDGES_PER_BLOCK];
    __shared__ long long s_src[EDGES_PER_BLOCK];

    const int e0 = blockIdx.x * EDGES_PER_BLOCK;
    const int t  = threadIdx.x;

    // Wave 0 (lanes 0..31, full wave32) stages this block's int64 edge indices
    // into LDS with CDNA5 async global->LDS copies (tracked by ASYNCcnt).
    if (t < EDGES_PER_BLOCK) {
        const int e = e0 + t;
        if (e < nEdges) {
            unsigned lds_d = (unsigned)(size_t)(&s_dst[t]);
            unsigned lds_s = (unsigned)(size_t)(&s_src[t]);
            const long long* gd = e_dst + e;
            const long long* gs = e_src + e;
            asm volatile("global_load_async_to_lds_b64 %0, %1, off"
                         :: "v"(lds_d), "v"(gd) : "memory");
            asm volatile("global_load_async_to_lds_b64 %0, %1, off"
                         :: "v"(lds_s), "v"(gs) : "memory");
        }
        asm volatile("s_wait_asynccnt 0" ::: "memory");
    }
    __syncthreads();

    // 8 lanes per edge; lane k handles features [4k, 4k+4).
    const int le = t >> 3;
    const int e  = e0 + le;
    if (e >= nEdges) return;
    const int fo = (t & 7) * 4;

    const long long srcn = s_src[le];
    const long long dstn = s_dst[le];

    // One global_load_b128 per lane: a contiguous 128B row per edge (L2-resident x).
    const float4 v = *reinterpret_cast<const float4*>(x + srcn * D_FEAT + fo);

    float* o = out + dstn * D_FEAT + fo;
    // Native f32 fadd atomics (global_atomic_add_f32), not a CAS loop.
    unsafeAtomicAdd(o + 0, v.x);
    unsafeAtomicAdd(o + 1, v.y);
    unsafeAtomicAdd(o + 2, v.z);
    unsafeAtomicAdd(o + 3, v.w);
}

__global__ __launch_bounds__(256) void zero_out_kernel(float4* __restrict__ out, int n4) {
    int i = blockIdx.x * blockDim.x + threadIdx.x;
    if (i < n4) out[i] = make_float4(0.f, 0.f, 0.f, 0.f);
}

extern "C" void kernel_launch(void* const* d_in, const int* in_sizes, int n_in,
                              void* d_out, int out_size, void* d_ws, size_t ws_size,
                              hipStream_t stream) {
    const float* x = (const float*)d_in[0];
    const long long* en = (const long long*)d_in[1];
    const int nEdges = in_sizes[1] / 2;
    const long long* e_dst = en;            // row 0: segment ids (destinations)
    const long long* e_src = en + nEdges;   // row 1: gather sources
    float* out = (float*)d_out;

    // d_out is poisoned by the harness; zero it first (float 0.0 == all-zero bytes).
    const int n4 = out_size / 4;
    zero_out_kernel<<<(n4 + 255) / 256, 256, 0, stream>>>((float4*)out, n4);

    const int blocks = (nEdges + EDGES_PER_BLOCK - 1) / EDGES_PER_BLOCK;
    scatter_add_kernel<<<blocks, BLOCK_THREADS, 0, stream>>>(x, e_dst, e_src, out, nEdges);
}